// DiffKS_10093173145696
// MI455X (gfx1250) — compile-verified
//
#include <hip/hip_runtime.h>
#include <hip/hip_bf16.h>
#include <math.h>

typedef __attribute__((ext_vector_type(2))) float v2f;
typedef __attribute__((ext_vector_type(8))) float v8f;

#define NF    64      // frames (spline knots)
#define LORD  5       // filter order
#define RING  1024    // ring buffer (power of two), >= 406 + CHUNK
#define CHUNK 32      // parallel samples per step; safe since z_l+1 >= ~101

// Branchless emulation of searchsorted(linspace(0,1,64), t, 'right') - 1,
// clipped to [0,62]. Pure selects -> no EXEC churn around the WMMAs.
__device__ __forceinline__ int spline_idx(float t) {
    const float h = 1.0f / 63.0f;
    int j = (int)(t * 63.0f);
    j = j < 0 ? 0 : (j > 62 ? 62 : j);
    const float tj  = (float)j * h;
    const float tj1 = (float)(j + 1) * h;
    const int jdown = j - 1;
    const int jup   = (j < 62 && t >= tj1) ? j + 1 : j;
    j = (t < tj) ? jdown : jup;
    j = j < 0 ? 0 : (j > 62 ? 62 : j);
    return j;
}

// ---------------------------------------------------------------------------
// Kernel 1: coefficient constraint (sigmoid + normalize) and natural-cubic-
// spline second derivatives via Thomas solve. 6 channels: [delay, b0..b4].
// Outputs: Yk[64*6] knot values, Msec[64*6] second derivatives (zero ends).
// ---------------------------------------------------------------------------
__global__ void diffks_prep(const float* __restrict__ delay,
                            const float* __restrict__ raw,
                            float* __restrict__ Yk,
                            float* __restrict__ Msec) {
    __shared__ float sY[NF * 6];
    const int f = threadIdx.x;
    if (f < NF) {
        sY[f * 6 + 0] = delay[f];
        float sg[LORD];
        float ssum = 0.0f;
#pragma unroll
        for (int j = 0; j < LORD; ++j) {
            float v = 1.0f / (1.0f + expf(-raw[f * LORD + j]));
            sg[j] = v;
            ssum += v;
        }
        const float inv = 1.0f / ssum;
#pragma unroll
        for (int j = 0; j < LORD; ++j) sY[f * 6 + 1 + j] = sg[j] * inv;
    }
    __syncthreads();
    if (f < NF) {
#pragma unroll
        for (int c = 0; c < 6; ++c) Yk[f * 6 + c] = sY[f * 6 + c];
    }
    if (f < 6) {                     // one channel per thread: Thomas algorithm
        const float h    = 1.0f / 63.0f;
        const float diag = 4.0f * h; // 2*(h+h), off-diag = h (uniform knots)
        const float rs   = 6.0f / h; // rhs scale
        float cp[62], dp[62];
        float y0 = sY[0 * 6 + f], y1 = sY[1 * 6 + f], y2 = sY[2 * 6 + f];
        float r = rs * (y2 - 2.0f * y1 + y0);
        float m = diag;
        cp[0] = h / m;
        dp[0] = r / m;
        for (int i = 1; i < 62; ++i) {
            y0 = y1; y1 = y2; y2 = sY[(i + 2) * 6 + f];
            r = rs * (y2 - 2.0f * y1 + y0);
            m = diag - h * cp[i - 1];
            cp[i] = h / m;
            dp[i] = (r - h * dp[i - 1]) / m;
        }
        float xi = dp[61];
        Msec[62 * 6 + f] = xi;
        for (int i = 60; i >= 0; --i) {
            xi = dp[i] - cp[i] * xi;
            Msec[(i + 1) * 6 + f] = xi;
        }
        Msec[0 * 6 + f]  = 0.0f;     // natural boundary conditions
        Msec[63 * 6 + f] = 0.0f;
    }
}

// ---------------------------------------------------------------------------
// Kernel 2: spline evaluation via V_WMMA_F32_16X16X4_F32.
// Per wave: one tile of 16 consecutive samples.
//   A[16x4]  row m = [1, dt_m, dt_m^2, dt_m^3]
//   B[4x16]  col c = cubic coeffs [p0,p1,p2,p3] of channel c (6 used)
//   D = A x B -> D[m][c] = interpolated channel c at sample m.
// Two WMMAs (interval j0 and j0+1) + one float-compare per row handle the
// knot straddle. Then 128 threads/block emit vals[T*8] (sparse row + z + x).
// ---------------------------------------------------------------------------
__global__ void diffks_spline_wmma(const float* __restrict__ Yk,
                                   const float* __restrict__ Msec,
                                   const float* __restrict__ exc,
                                   int exc_len,
                                   float* __restrict__ vals,
                                   int T) {
    __shared__ float smem[128 * 6];
    const int lane = threadIdx.x & 31;
    const int wave = threadIdx.x >> 5;
    const int tile = blockIdx.x * 8 + wave;
    const int tile_base = tile * 16;

    const float h   = 1.0f / 63.0f;
    const float inv = 1.0f / (float)(T - 1);
    const bool  hi  = lane >= 16;
    const int   row = lane & 15;

    int s = tile_base + row;
    if (s >= T) s = T - 1;
    const float t = (float)s * inv;

    // interval of the tile's first sample (uniform across the wave)
    const float t0 = (float)(tile_base < T ? tile_base : T - 1) * inv;
    const int j0 = spline_idx(t0);
    const int j1 = j0 < 62 ? j0 + 1 : 62;
    const float tknot1 = (float)(j0 + 1) * h;   // boundary for row-wise select
    const bool  can_adv = (j0 < 62);

    // A matrices (32-bit A 16x4 layout: VGPR0 holds K=0|2, VGPR1 holds K=1|3)
    const float dta = t - (float)j0 * h;
    const float dtb = t - (float)j1 * h;
    v2f Aa, Ab;
    Aa.x = hi ? dta * dta : 1.0f;  Aa.y = hi ? dta * dta * dta : dta;
    Ab.x = hi ? dtb * dtb : 1.0f;  Ab.y = hi ? dtb * dtb * dtb : dtb;

    // B matrices (4x16, col = channel; VGPR0 holds K=0|2, VGPR1 holds K=1|3)
    const int c  = lane & 15;
    const int cc = c < 5 ? c : 5;   // clamp unused columns to a valid channel
    v2f Ba, Bb;
    {
        float yj  = Yk[j0 * 6 + cc],   yj1 = Yk[(j0 + 1) * 6 + cc];
        float mj  = Msec[j0 * 6 + cc], mj1 = Msec[(j0 + 1) * 6 + cc];
        float p0 = yj;
        float p1 = (yj1 - yj) * 63.0f - (2.0f * mj + mj1) * (h / 6.0f);
        float p2 = 0.5f * mj;
        float p3 = (mj1 - mj) * 10.5f;            // (M[j+1]-M[j])/(6h)
        Ba.x = hi ? p2 : p0;  Ba.y = hi ? p3 : p1;
    }
    {
        float yj  = Yk[j1 * 6 + cc],   yj1 = Yk[(j1 + 1) * 6 + cc];   // j1+1<=63
        float mj  = Msec[j1 * 6 + cc], mj1 = Msec[(j1 + 1) * 6 + cc];
        float p0 = yj;
        float p1 = (yj1 - yj) * 63.0f - (2.0f * mj + mj1) * (h / 6.0f);
        float p2 = 0.5f * mj;
        float p3 = (mj1 - mj) * 10.5f;
        Bb.x = hi ? p2 : p0;  Bb.y = hi ? p3 : p1;
    }

    v8f Cz = {};
    // (neg_a, A, neg_b, B, c_mod, C, reuse_a, reuse_b) ; EXEC is all-ones here
    v8f Da = __builtin_amdgcn_wmma_f32_16x16x4_f32(false, Aa, false, Ba,
                                                   (short)0, Cz, false, false);
    v8f Db = __builtin_amdgcn_wmma_f32_16x16x4_f32(false, Ab, false, Bb,
                                                   (short)0, Cz, false, false);

    // Scatter D to LDS: VGPR r holds row (r + 8*hi), column = lane&15.
    // Row uses interval j0+1 iff its t is past the knot (monotone, <=1 step).
    if (c < 6) {
#pragma unroll
        for (int r = 0; r < 8; ++r) {
            const int rrow = r + (hi ? 8 : 0);
            int sr = tile_base + rrow;
            if (sr >= T) sr = T - 1;
            const float tr    = (float)sr * inv;
            const bool  use_b = can_adv && (tr >= tknot1);
            const float v     = use_b ? Db[r] : Da[r];
            smem[(wave * 16 + rrow) * 6 + c] = v;
        }
    }
    __syncthreads();

    // Phase 2: 128 samples/block -> sparse row values, z, and excitation x
    const int i = threadIdx.x;
    if (i < 128) {
        const int sg = blockIdx.x * 128 + i;
        if (sg < T) {
            const float dly = smem[i * 6 + 0];
            const float b0 = smem[i * 6 + 1], b1 = smem[i * 6 + 2];
            const float b2 = smem[i * 6 + 3], b3 = smem[i * 6 + 4];
            const float b4 = smem[i * 6 + 5];
            const float zf   = floorf(dly);
            const float alfa = dly - zf;
            const float om   = 1.0f - alfa;
            const float xv   = (sg < exc_len) ? exc[sg] : 0.0f;
            float4 vA = make_float4(-om * b0,
                                    -(alfa * b0 + om * b1),
                                    -(alfa * b1 + om * b2),
                                    -(alfa * b2 + om * b3));
            float4 vB = make_float4(-(alfa * b3 + om * b4),
                                    -alfa * b4,
                                    zf, xv);
            *(float4*)(vals + (size_t)sg * 8 + 0) = vA;
            *(float4*)(vals + (size_t)sg * 8 + 4) = vB;
        }
    }
}

// ---------------------------------------------------------------------------
// Kernel 3: sequential recurrence. One wave32; chunks of 32 samples are
// independent because the dependency distance is z_l+1 >= ~101. History kept
// in an LDS ring buffer (negative indices wrap to zero-initialized slots).
// Software-pipelined: next chunk's coefficients preload into registers while
// the current chunk computes; prefetch pulls lines two chunks ahead.
// ---------------------------------------------------------------------------
__global__ void diffks_recur(const float* __restrict__ vals,
                             float* __restrict__ out,
                             int T) {
    __shared__ float ring[RING];
    const int lane = threadIdx.x;
    for (int i = lane; i < RING; i += CHUNK) ring[i] = 0.0f;
    __syncthreads();

    const int nchunk = T / CHUNK;
    float4 vA = *(const float4*)(vals + (size_t)lane * 8 + 0);
    float4 vB = *(const float4*)(vals + (size_t)lane * 8 + 4);

    for (int ch = 0; ch < nchunk; ++ch) {
        const int t = ch * CHUNK + lane;

        // preload next chunk while this chunk's LDS gather/FMA chain runs
        float4 nA = vA, nB = vB;
        if (ch + 1 < nchunk) {
            const float* p = vals + (size_t)(t + CHUNK) * 8;
            nA = *(const float4*)(p + 0);
            nB = *(const float4*)(p + 4);
        }
        if (ch + 2 < nchunk) {
            __builtin_prefetch(vals + (size_t)(t + 2 * CHUNK) * 8, 0, 1);
        }

        const int z = (int)vB.z;
        const int p = t - 1 - z;            // reads are all < ch*CHUNK
        float acc = vB.w;                   // x[t] baked in by kernel 2
        acc -= vA.x * ring[(p    ) & (RING - 1)];
        acc -= vA.y * ring[(p - 1) & (RING - 1)];
        acc -= vA.z * ring[(p - 2) & (RING - 1)];
        acc -= vA.w * ring[(p - 3) & (RING - 1)];
        acc -= vB.x * ring[(p - 4) & (RING - 1)];
        acc -= vB.y * ring[(p - 5) & (RING - 1)];
        ring[t & (RING - 1)] = acc;         // disjoint from this chunk's reads
        out[t] = acc;
        vA = nA; vB = nB;
        __syncthreads();                    // order writes before next reads
    }
}

extern "C" void kernel_launch(void* const* d_in, const int* in_sizes, int n_in,
                              void* d_out, int out_size, void* d_ws, size_t ws_size,
                              hipStream_t stream) {
    const float* delay = (const float*)d_in[0];
    const float* raw   = (const float*)d_in[1];
    const float* exc   = (const float*)d_in[2];
    const int exc_len  = in_sizes[2];
    const int T        = out_size;          // n_samples (65536)

    float* Yk   = (float*)d_ws;                         // 64*6 floats
    float* Msec = Yk + NF * 6;                          // 64*6 floats
    float* vals = (float*)((char*)d_ws + 8192);         // T*8 floats, 16B aligned

    diffks_prep<<<1, 64, 0, stream>>>(delay, raw, Yk, Msec);

    const int blocks = (T + 127) / 128;                 // 128 samples per block
    diffks_spline_wmma<<<blocks, 256, 0, stream>>>(Yk, Msec, exc, exc_len, vals, T);

    diffks_recur<<<1, CHUNK, 0, stream>>>(vals, (float*)d_out, T);
}